// LocallyConnected3DFlipout_81123342287365
// MI455X (gfx1250) — compile-verified
//
#include <hip/hip_runtime.h>

// ---- problem constants (match the reference) ----
#define IXX       32
#define IYY       32
#define IZZ       16
#define CINN      8
#define FF        8
#define OXX       30
#define OYY       30
#define OZZ       14
#define IN_SIZE   (IXX*IYY*IZZ*CINN)   // 131072
#define OUT_SIZE  (OXX*OYY*OZZ*FF)     // 100800
#define KTOT      (3*3*3*CINN)         // 216 weights per (site, filter)
#define WPS       (FF*KTOT)            // 1728 weights per site per array
#define NSPAT     (OXX*OYY*OZZ)        // 12600
#define SPW       2                    // spatial sites per wave (exact staging)

typedef float v2f __attribute__((ext_vector_type(2)));
typedef float v4f __attribute__((ext_vector_type(4)));
typedef float v8f __attribute__((ext_vector_type(8)));

// voxel -> input-flat offset: VOX[t] = kx*4096 + ky*128 + kz*8, t=(kx*3+ky)*3+kz
__device__ __constant__ const int VOX[27] = {
       0,    8,   16,  128,  136,  144,  256,  264,  272,
    4096, 4104, 4112, 4224, 4232, 4240, 4352, 4360, 4368,
    8192, 8200, 8208, 8320, 8328, 8336, 8448, 8456, 8464
};

__global__ __launch_bounds__(32)
void lc3d_flipout_wmma(const float* __restrict__ x,        // [B, IN_SIZE]
                       const float* __restrict__ kmean,    // [NNZ]
                       const float* __restrict__ kstd,     // [NNZ]
                       const float* __restrict__ bias,     // [OUT_SIZE]
                       const float* __restrict__ eps,      // [NNZ]
                       const float* __restrict__ sign_in,  // [B, IN_SIZE]
                       const float* __restrict__ sign_out, // [B, OUT_SIZE]
                       float* __restrict__ out)            // [B, OUT_SIZE]
{
    // LDS: [0, SPW*WPS) = kernel_mean rows, [SPW*WPS, 2*SPW*WPS) = std*eps rows
    __shared__ float lds[2 * SPW * WPS];                   // 27648 B
    const int lane  = threadIdx.x;                         // wave32
    const int spat0 = blockIdx.x * SPW;

    // ---------------- stage weights (the 261 MB read-once stream) ----------------
    // 2 sites * 1728 floats per array = 27 * (32 lanes * float4): exact, coalesced.
    const v4f* gm = (const v4f*)(kmean + (long long)spat0 * WPS);
    const v4f* gs = (const v4f*)(kstd  + (long long)spat0 * WPS);
    const v4f* ge = (const v4f*)(eps   + (long long)spat0 * WPS);
    v4f* lm = (v4f*)(lds);
    v4f* lp = (v4f*)(lds + SPW * WPS);
    #pragma unroll 3
    for (int i = 0; i < 27; ++i) {
        const int j = i * 32 + lane;
        v4f m = __builtin_nontemporal_load(gm + j);        // TH=NT: don't pollute L2
        v4f s = __builtin_nontemporal_load(gs + j);
        v4f e = __builtin_nontemporal_load(ge + j);
        lm[j] = m;
        lp[j] = s * e;                                     // fuse std*eps at stage time
    }
    __syncthreads();

    // ---------------- per-lane WMMA fragment roles ----------------
    // A (16x4 f32): lanes 0-15 rows M=0..15, K=kbase+{0,1}; lanes 16-31 K=kbase+{2,3}.
    // Rows 0-7 = W_mean[f], rows 8-15 = (W_std*eps)[f].
    const int m  = lane & 15;
    const int kp = lane >> 4;                              // K offset 0 / 2
    const int c  = 2 * kp;                                 // lane-constant ci (half adds +4)
    const int arow = (m < 8) ? (m * KTOT) : (SPW * WPS + (m - 8) * KTOT);
    // B (4x16 f32): column n = lane&15.  n<4: x[b]; 4<=n<8: x[b]*sign_in[b]; n>=8: 0.
    const int   bidx    = m & 3;
    const bool  bSigned = (m >= 4) && (m < 8);
    const float maskc   = (m < 8) ? 1.0f : 0.0f;           // 0 for unused columns
    const float* xrow = x       + (size_t)bidx * IN_SIZE;
    const float* srow = sign_in + (size_t)bidx * IN_SIZE;

    for (int s = 0; s < SPW; ++s) {
        const int spat = spat0 + s;
        const int oz  = spat % OZZ;
        const int oxy = spat / OZZ;
        const int oy  = oxy % OYY;
        const int ox  = oxy / OYY;
        const int ibase = ox * (IYY * IZZ * CINN) + oy * (IZZ * CINN) + oz * CINN;

        const float* aP = lds + arow + s * WPS + c;        // per-lane LDS base
        const float* xb = xrow + (ibase + c);              // per-lane global bases
        const float* sb = srow + (ibase + c);              // (always-valid addresses)

        v8f acc0 = {}, acc1 = {};
        // k0 = 8*i + c (+4 for half1): voxel index t == i, so every access is
        // base + compile-time immediate offset. Branch-free; EXEC stays all-ones.
        #pragma unroll
        for (int i = 0; i < 27; ++i) {
            const int vo = VOX[i];
            v2f a0 = *(const v2f*)(aP + 8 * i);            // ds_load_b64 offset:32i
            v2f a1 = *(const v2f*)(aP + 8 * i + 4);        // ds_load_b64 offset:32i+16
            v2f x0 = *(const v2f*)(xb + vo);               // global_load_b64 offset:imm
            v2f x1 = *(const v2f*)(xb + vo + 4);
            v2f s0 = *(const v2f*)(sb + vo);
            v2f s1 = *(const v2f*)(sb + vo + 4);
            v2f m0, m1;                                    // per-element select, no branches
            m0.x = bSigned ? s0.x : maskc;
            m0.y = bSigned ? s0.y : maskc;
            m1.x = bSigned ? s1.x : maskc;
            m1.y = bSigned ? s1.y : maskc;
            v2f b0 = x0 * m0;
            v2f b1 = x1 * m1;
            acc0 = __builtin_amdgcn_wmma_f32_16x16x4_f32(
                false, a0, false, b0, (short)0, acc0, false, false);
            acc1 = __builtin_amdgcn_wmma_f32_16x16x4_f32(
                false, a1, false, b1, (short)0, acc1, false, false);
        }
        v8f acc = acc0 + acc1;

        // D layout: VGPR r, lanes 0-15 -> (M=r, N=lane); lanes 16-31 -> (M=r+8, N=lane-16).
        // mean[f][b] at lanes 0..3 elem f;  pert[f][b] at lanes 20..23 elem f.
        float pf[8];
        #pragma unroll
        for (int r = 0; r < 8; ++r)
            pf[r] = __shfl(acc[r], 20 + (lane & 3), 32);

        if (lane < 4) {
            const int b = lane;
            const long long row0 = (long long)spat * FF;
            const float* so = sign_out + (size_t)b * OUT_SIZE + row0;
            const float* bi = bias + row0;
            float*       op = out  + (size_t)b * OUT_SIZE + row0;
            v4f lo, hi;
            #pragma unroll
            for (int r = 0; r < 4; ++r) {
                lo[r] = acc[r]     + pf[r]     * so[r]     + bi[r];
                hi[r] = acc[r + 4] + pf[r + 4] * so[r + 4] + bi[r + 4];
            }
            *(v4f*)(op)     = lo;                          // 16B aligned (spat*32 B)
            *(v4f*)(op + 4) = hi;
        }
    }
}

extern "C" void kernel_launch(void* const* d_in, const int* in_sizes, int n_in,
                              void* d_out, int out_size, void* d_ws, size_t ws_size,
                              hipStream_t stream) {
    // setup_inputs order: inputs, kernel_mean, kernel_std, bias, eps, sign_in,
    //                     sign_out, rows, cols   (rows/cols unused: closed-form indexing)
    const float* x        = (const float*)d_in[0];
    const float* kmean    = (const float*)d_in[1];
    const float* kstd     = (const float*)d_in[2];
    const float* bias     = (const float*)d_in[3];
    const float* eps      = (const float*)d_in[4];
    const float* sign_in  = (const float*)d_in[5];
    const float* sign_out = (const float*)d_in[6];
    float* out = (float*)d_out;

    const int blocks = NSPAT / SPW;                        // 6300 one-wave workgroups
    lc3d_flipout_wmma<<<blocks, 32, 0, stream>>>(
        x, kmean, kstd, bias, eps, sign_in, sign_out, out);
}